// NetworkBasic_21139829031401
// MI455X (gfx1250) — compile-verified
//
#include <hip/hip_runtime.h>
#include <hip/hip_bf16.h>

#define B_ 8
#define C_ 2
#define H_ 17
#define W_ 17
#define T_ 350
#define O_ 8
#define NT_ 22   // ceil(350/16) time tiles for WMMA conv

typedef float v2f __attribute__((ext_vector_type(2)));
typedef float v8f __attribute__((ext_vector_type(8)));

__device__ __forceinline__ int clampi(int v, int lo, int hi) {
    return v < lo ? lo : (v > hi ? hi : v);
}

// Tap descriptor for conv feature k at output position (h,w).  When k is a
// compile-time constant (unrolled loop), all div/mod folds to constants and
// off becomes SALU arithmetic on the wave-uniform h,w.
__device__ __forceinline__ void conv_tap(int k, int h, int w, int& off, bool& ok) {
    int kk = k < 49 ? k : 49;               // memory-safe clamp for pad taps
    int c  = kk / 25, rr = kk % 25;
    int y  = h + rr / 5 - 2;
    int x  = w + rr % 5 - 2;
    ok  = (k < 50) && (y >= 0) && (y < H_) && (x >= 0) && (x < W_);
    off = (c * H_ + clampi(y, 0, H_ - 1)) * W_ + clampi(x, 0, W_ - 1);
}

// ---------------------------------------------------------------------------
// K1: alpha_psp of the input (tau=1).  rho=e^-1, C*rho = 1 exactly.
// One wave owns 32 pixels; T in 32-step chunks staged through LDS so global
// traffic is coalesced; stride-33 rows -> bank-conflict-free scan.
// ---------------------------------------------------------------------------
__global__ __launch_bounds__(256) void k_alpha_psp_in(const float* __restrict__ x,
                                                      float* __restrict__ psp1) {
    __shared__ float tile[8][32][33];
    const int lane = threadIdx.x & 31;
    const int wv   = threadIdx.x >> 5;
    const int pixBase = (blockIdx.x * 8 + wv) * 32;
    const int NPIX = B_ * C_ * H_ * W_;
    const bool allpix = (pixBase + 32 <= NPIX);
    const float rho = 0.36787944117144233f;    // exp(-1)
    float u = 0.f, p = 0.f;
    for (int tc = 0; tc < T_; tc += 32) {
        const bool full = allpix && (tc + 32 <= T_);
        if (full) {
#pragma unroll 4
            for (int pp = 0; pp < 32; ++pp)    // coalesced: lane = time offset
                tile[wv][pp][lane] = x[(pixBase + pp) * T_ + tc + lane];
        } else {
            for (int pp = 0; pp < 32; ++pp) {
                int pix = pixBase + pp, t = tc + lane;
                tile[wv][pp][lane] = (pix < NPIX && t < T_) ? x[pix * T_ + t] : 0.f;
            }
        }
        __syncthreads();
#pragma unroll
        for (int j = 0; j < 32; ++j) {         // lane scans its own pixel
            float s = tile[wv][lane][j];
            p = rho * p + u;                   // C*rho == 1 for tau=1
            u = rho * u + s;
            tile[wv][lane][j] = p;
        }
        __syncthreads();
        if (full) {
#pragma unroll 4
            for (int pp = 0; pp < 32; ++pp)
                psp1[(pixBase + pp) * T_ + tc + lane] = tile[wv][pp][lane];
        } else {
            for (int pp = 0; pp < 32; ++pp) {
                int pix = pixBase + pp, t = tc + lane;
                if (pix < NPIX && t < T_) psp1[pix * T_ + t] = tile[wv][pp][lane];
            }
        }
        __syncthreads();
    }
}

// ---------------------------------------------------------------------------
// K2: 5x5 conv (2->8 ch, pad 2) as fp32 WMMA GEMM per (b,h,w,t-tile):
//   D[o,t] += A[o,k] * B[k,t], k=(c,ky,kx) in [0,50) padded to 52,
//   13 x V_WMMA_F32_16X16X4_F32 per tile.
// job is wave-uniform (readfirstlane -> SALU decomposition).  Each unrolled
// B-tap has exactly two possible k values (lane-half kb in {0,2}); both tap
// descriptors fold to compile-time constants + uniform h,w arithmetic, so
// per-tap cost is ~1 cndmask + 1 load.  All loads use clamped memory-safe
// addresses with a zero-select (no exec-mask manipulation inside the wave,
// keeping EXEC all-ones as WMMA requires).
// ---------------------------------------------------------------------------
__global__ __launch_bounds__(256) void k_conv_wmma(const float* __restrict__ psp1,
                                                   const float* __restrict__ cw,
                                                   float* __restrict__ wpsp) {
    const int jobs = B_ * H_ * W_ * NT_;
    const int lane = threadIdx.x & 31;
    const int wv   = threadIdx.x >> 5;
    int job = __builtin_amdgcn_readfirstlane(blockIdx.x * 8 + wv);
    if (job >= jobs) return;                   // uniform per wave
    int tt = job % NT_; int r = job / NT_;     // SALU (uniform)
    int w  = r % W_;  r /= W_;
    int h  = r % H_;  int b = r / H_;
    const int t0 = tt * 16;
    const int m  = lane & 15;                  // A row (output channel)
    const bool hiK = (lane & 16) != 0;         // lane half selects K sub-rows
    const int kb = hiK ? 2 : 0;
    // Preload A fragments (weights).  Branchless clamp + select.
    v2f A[13];
#pragma unroll
    for (int ks = 0; ks < 13; ++ks) {
        int k0 = ks * 4 + kb;
        int ia = clampi(m * 50 + k0, 0, O_ * 50 - 2);
        float w0 = cw[ia], w1 = cw[ia + 1];
        bool vm = (m < 8);
        A[ks].x = (vm && k0     < 50) ? w0 : 0.f;
        A[ks].y = (vm && k0 + 1 < 50) ? w1 : 0.f;
    }
    const int  t   = t0 + m;                   // B column = timestep (per lane)
    const bool tOK = (t < T_);
    const int  tcl = tOK ? t : T_ - 1;         // clamped memory-safe timestep
    const float* pb = psp1 + (size_t)(b * C_ * H_ * W_) * T_;
    v8f acc = {};
#pragma unroll
    for (int ks = 0; ks < 13; ++ks) {
        v2f Bf;
#pragma unroll
        for (int q = 0; q < 2; ++q) {
            int offLo, offHi; bool okLo, okHi;
            conv_tap(ks * 4 + q,     h, w, offLo, okLo);   // lanes 0..15
            conv_tap(ks * 4 + 2 + q, h, w, offHi, okHi);   // lanes 16..31
            int  off = hiK ? offHi : offLo;                // one cndmask
            bool ok  = (hiK ? okHi : okLo) && tOK;
            float v  = pb[(size_t)off * T_ + tcl];
            v = ok ? v : 0.f;
            if (q == 0) Bf.x = v; else Bf.y = v;
        }
        acc = __builtin_amdgcn_wmma_f32_16x16x4_f32(
            /*neg_a=*/false, A[ks], /*neg_b=*/false, Bf,
            /*c_mod=*/(short)0, acc, /*reuse_a=*/false, /*reuse_b=*/false);
    }
    // D layout: lanes 0..15 hold rows M=0..7 in vgprs 0..7 -> exactly o=0..7
    if (lane < 16 && tOK) {
#pragma unroll
        for (int i = 0; i < 8; ++i)
            wpsp[(((b * O_ + i) * H_ + h) * W_ + w) * T_ + t] = acc[i];
    }
}

// ---------------------------------------------------------------------------
// K3: spike_dynamics stage 1 (theta=30, tauRef=1) fused with alpha_psp(tau=4)
// of the resulting spikes -> psp2 (s1 never materialized).  Same LDS-staged
// coalesced scan structure as K1; 4 IIR states per pixel held in registers.
// 18496 pixels = 578 full waves, so the pixel fast path always hits.
// ---------------------------------------------------------------------------
__global__ __launch_bounds__(256) void k_dyn1_psp2(const float* __restrict__ wpsp,
                                                   float* __restrict__ psp2) {
    __shared__ float tile[8][32][33];
    const int lane = threadIdx.x & 31;
    const int wv   = threadIdx.x >> 5;
    const int pixBase = (blockIdx.x * 8 + wv) * 32;
    const int NPIX = B_ * O_ * H_ * W_;
    const bool allpix = (pixBase + 32 <= NPIX);
    const float rhoR   = 0.36787944117144233f;  // exp(-1)
    const float CRrho  = -60.0f;                // (-2*30*e)*exp(-1)
    const float rho2   = 0.77880078307140487f;  // exp(-0.25)
    const float C2rho2 = 0.52925000945116479f;  // exp(0.75)/4
    float us = 0.f, rr = 0.f, ua = 0.f, pa = 0.f;
    for (int tc = 0; tc < T_; tc += 32) {
        const bool full = allpix && (tc + 32 <= T_);
        if (full) {
#pragma unroll 4
            for (int pp = 0; pp < 32; ++pp)
                tile[wv][pp][lane] = wpsp[(pixBase + pp) * T_ + tc + lane];
        } else {
            for (int pp = 0; pp < 32; ++pp) {
                int pix = pixBase + pp, t = tc + lane;
                tile[wv][pp][lane] = (pix < NPIX && t < T_) ? wpsp[pix * T_ + t] : 0.f;
            }
        }
        __syncthreads();
#pragma unroll
        for (int j = 0; j < 32; ++j) {
            float win = tile[wv][lane][j];
            rr = rhoR * rr + CRrho * us;                    // refractory from old u
            float s = (win + rr >= 30.0f) ? 1.0f : 0.0f;    // spike
            us = rhoR * us + s;
            pa = rho2 * pa + C2rho2 * ua;                   // alpha PSP of spikes
            ua = rho2 * ua + s;
            tile[wv][lane][j] = pa;
        }
        __syncthreads();
        if (full) {
#pragma unroll 4
            for (int pp = 0; pp < 32; ++pp)
                psp2[(pixBase + pp) * T_ + tc + lane] = tile[wv][pp][lane];
        } else {
            for (int pp = 0; pp < 32; ++pp) {
                int pix = pixBase + pp, t = tc + lane;
                if (pix < NPIX && t < T_) psp2[pix * T_ + t] = tile[wv][pp][lane];
            }
        }
        __syncthreads();
    }
}

// ---------------------------------------------------------------------------
// K4: fused convTranspose4 (non-overlapping: each 68-grid pixel maps to one
// psp2 source pixel) + bilinear-4x skip + spike_dynamics stage 2 + final
// 68->34 bilinear (== exact 2x2 average with half-pixel centers).
// Each thread owns one output pixel = a 2x2 s2 block (4 recurrences in regs);
// u2 and s2 (103 MB each) are never materialized.
// ---------------------------------------------------------------------------
__global__ __launch_bounds__(256) void k_fuse_out(const float* __restrict__ psp1,
                                                  const float* __restrict__ psp2,
                                                  const float* __restrict__ uw,
                                                  float* __restrict__ out) {
    int idx = blockIdx.x * 256 + threadIdx.x;
    const int NOUT = B_ * C_ * 34 * 34;
    if (idx >= NOUT) return;
    int X = idx % 34; int rem = idx / 34;
    int Y = rem % 34; rem /= 34;
    int co = rem % C_; int b = rem / C_;
    const int hy = Y >> 1, hx = X >> 1;        // (2Y)/4 == (2Y+1)/4 == Y>>1
    // convT weights for the 4 block positions (uw shape [Ci=8][Co=2][4][4])
    float wblk[2][2][8];
    int iy0 = (2 * Y) & 3, ix0 = (2 * X) & 3;
    for (int py = 0; py < 2; ++py)
        for (int px = 0; px < 2; ++px)
            for (int ci = 0; ci < 8; ++ci)
                wblk[py][px][ci] = uw[((ci * C_ + co) * 4 + (iy0 + py)) * 4 + (ix0 + px)];
    // bilinear 4x upsample taps (half-pixel centers, edge clamp)
    int yi0[2], yi1[2], xi0[2], xi1[2]; float yf[2], xf[2];
    for (int p = 0; p < 2; ++p) {
        float sy = (2 * Y + p + 0.5f) * 0.25f - 0.5f;
        float fy = floorf(sy); yf[p] = sy - fy;
        yi0[p] = clampi((int)fy, 0, 16); yi1[p] = clampi((int)fy + 1, 0, 16);
        float sx = (2 * X + p + 0.5f) * 0.25f - 0.5f;
        float fx = floorf(sx); xf[p] = sx - fx;
        xi0[p] = clampi((int)fx, 0, 16); xi1[p] = clampi((int)fx + 1, 0, 16);
    }
    const float rhoR  = 0.77880078307140487f;   // exp(-0.25)
    const float CRrho = -105.85000189023297f;   // -50*exp(0.75)
    const long pspc  = (long)(b * C_ + co) * H_ * W_;
    const long psp2b = (long)b * O_ * H_ * W_ + hy * W_ + hx;
    float u[4] = {0, 0, 0, 0}, rf[4] = {0, 0, 0, 0};
    float* op = out + (long)idx * T_;
    for (int t = 0; t < T_; ++t) {
        float p2[8];
#pragma unroll
        for (int ci = 0; ci < 8; ++ci)
            p2[ci] = psp2[(psp2b + (long)ci * H_ * W_) * T_ + t];
        float accs = 0.f;
#pragma unroll
        for (int py = 0; py < 2; ++py) {
#pragma unroll
            for (int px = 0; px < 2; ++px) {
                float v00 = psp1[(pspc + yi0[py] * W_ + xi0[px]) * T_ + t];
                float v01 = psp1[(pspc + yi0[py] * W_ + xi1[px]) * T_ + t];
                float v10 = psp1[(pspc + yi1[py] * W_ + xi0[px]) * T_ + t];
                float v11 = psp1[(pspc + yi1[py] * W_ + xi1[px]) * T_ + t];
                float skip = (1.f - yf[py]) * ((1.f - xf[px]) * v00 + xf[px] * v01)
                           + yf[py]        * ((1.f - xf[px]) * v10 + xf[px] * v11);
                float u2v = skip;
#pragma unroll
                for (int ci = 0; ci < 8; ++ci) u2v += p2[ci] * wblk[py][px][ci];
                int q = py * 2 + px;
                rf[q] = rhoR * rf[q] + CRrho * u[q];
                float s = (u2v + rf[q] >= 100.0f) ? 1.0f : 0.0f;
                u[q] = rhoR * u[q] + s;
                accs += s;
            }
        }
        op[t] = 0.25f * accs;                   // 68->34 bilinear == 2x2 mean
    }
}

extern "C" void kernel_launch(void* const* d_in, const int* in_sizes, int n_in,
                              void* d_out, int out_size, void* d_ws, size_t ws_size,
                              hipStream_t stream) {
    const float* x  = (const float*)d_in[0];  // spikeInput [8,2,17,17,350]
    const float* cw = (const float*)d_in[1];  // conv1_w   [8,2,5,5]
    const float* uw = (const float*)d_in[2];  // upconv1_w [8,2,4,4]
    float* out = (float*)d_out;               // [8,2,34,34,350]
    float* ws  = (float*)d_ws;
    float* psp1 = ws;                         // 1,618,400 floats ( 6.5 MB)
    float* wpsp = psp1 + 1618400;             // 6,473,600 floats (25.9 MB)
    float* psp2 = wpsp + 6473600;             // 6,473,600 floats (25.9 MB)
    (void)in_sizes; (void)n_in; (void)out_size; (void)ws_size;

    // K1: 4624 pixels, 32 per wave, 8 waves per block
    int w1 = (B_ * C_ * H_ * W_ + 31) / 32;
    k_alpha_psp_in<<<(w1 + 7) / 8, 256, 0, stream>>>(x, psp1);

    // K2: one wave per (b,h,w,t-tile) WMMA job
    int jobs = B_ * H_ * W_ * NT_;
    k_conv_wmma<<<(jobs + 7) / 8, 256, 0, stream>>>(psp1, cw, wpsp);

    // K3: 18,496 pixels, 32 per wave
    int w3 = (B_ * O_ * H_ * W_ + 31) / 32;
    k_dyn1_psp2<<<(w3 + 7) / 8, 256, 0, stream>>>(wpsp, psp2);

    // K4: one thread per final output pixel (18,496 threads)
    int nout = B_ * C_ * 34 * 34;
    k_fuse_out<<<(nout + 255) / 256, 256, 0, stream>>>(psp1, psp2, uw, out);
}